// DynamicNet_17695265259799
// MI455X (gfx1250) — compile-verified
//
#include <hip/hip_runtime.h>
#include <math.h>

// ---------------------------------------------------------------------------
// DynamicNet (66-node strictly-lower-triangular tanh DAG) on gfx1250.
// Off-diagonal node-block contributions via V_WMMA_F32_16X16X4_F32 (fp32,
// matches reference math); intra-block 16-step triangular recurrence via
// per-lane VALU with scalar-broadcast W from LDS. Wave32, 4 waves / WG.
// tanh: hardware V_TANH_F32 if the builtin exists (CDNA5 lists TANH as a
// TRANS op), else branch-free v_exp_f32 + v_rcp_f32 formulation.
// ---------------------------------------------------------------------------

typedef __attribute__((ext_vector_type(2))) float v2f;
typedef __attribute__((ext_vector_type(8))) float v8f;

#define N_NODES    66
#define WT_STRIDE  68          // words; even => aligned ds_load_b64 A-operands, conflict-free
#define WT_ROWS    80          // zero-padded so tail block (nodes 64..79) reads zeros
#define A_STRIDE   40          // words; half-lane K rows land in disjoint bank groups
#define B_OFF      (WT_ROWS * WT_STRIDE)            // bias vector
#define A_OFF0     (B_OFF + N_NODES)                // per-wave activation regions
#define A_REGION   (N_NODES * A_STRIDE)
#define SH_FLOATS  (A_OFF0 + 4 * A_REGION)          // = 5440+66+10560 = 16066 (~63 KB)

__device__ __forceinline__ float act_tanh(float x) {
#if __has_builtin(__builtin_amdgcn_tanhf)
  return __builtin_amdgcn_tanhf(x);                  // V_TANH_F32 (1 TRANS op)
#elif __has_builtin(__builtin_amdgcn_tanh_f32)
  return __builtin_amdgcn_tanh_f32(x);               // V_TANH_F32 (1 TRANS op)
#else
  // Branch-free fallback: t = exp(-2|x|) in (0,1] -> never overflows.
  const float ax = __builtin_fabsf(x);
  const float t  = __builtin_amdgcn_exp2f(ax * -2.8853900817779268f); // -2*log2(e)
  const float r  = __builtin_amdgcn_rcpf(1.0f + t);
  return __builtin_copysignf((1.0f - t) * r, x);
#endif
}

// One node block: WMMA over all previous nodes, then sequential triangular part.
// JBASE: first node of block; KPREV: # previous nodes (== JBASE); NLOC: live nodes.
template <int JBASE, int KPREV, int NLOC>
__device__ __forceinline__ void do_block(float* sh, const int aOff,
                                         const int lane, const int m, const int h,
                                         float (&areg)[16]) {
  v8f d0 = {0.f, 0.f, 0.f, 0.f, 0.f, 0.f, 0.f, 0.f};
  v8f d1 = {0.f, 0.f, 0.f, 0.f, 0.f, 0.f, 0.f, 0.f};
  const int awBase = (JBASE + m) * WT_STRIDE + 2 * h;  // Wt row = output node, +K
  const int bwBase = aOff + (2 * h) * A_STRIDE + m;    // activation K-row, col

#pragma unroll
  for (int i0 = 0; i0 < KPREV; i0 += 4) {
    // A = W^T tile chunk (16 j-rows x 4 i-cols): one aligned b64 per lane.
    v2f a = *(const v2f*)&sh[awBase + i0];
    // B = activations chunk (4 i-rows x 16 cols), two N-tiles.
    const float* bp = &sh[bwBase + i0 * A_STRIDE];
    v2f b0 = {bp[0],  bp[A_STRIDE]};
    v2f b1 = {bp[16], bp[A_STRIDE + 16]};
    d0 = __builtin_amdgcn_wmma_f32_16x16x4_f32(false, a, false, b0, (short)0, d0, false, false);
    d1 = __builtin_amdgcn_wmma_f32_16x16x4_f32(false, a, false, b1, (short)0, d1, false, false);
  }

  // Spill D tiles (partial sums) to the [node][col] activation region.
  // D layout: VGPR r, lanes 0-15 -> row r; lanes 16-31 -> row r+8.
#pragma unroll
  for (int r = 0; r < 8; ++r) {
    if (JBASE + r + 8 * h < N_NODES) {   // compile-time true except tail block
      sh[aOff + (JBASE + r + 8 * h) * A_STRIDE + m]      = d0[r];
      sh[aOff + (JBASE + r + 8 * h) * A_STRIDE + m + 16] = d1[r];
    }
  }
  asm volatile("s_wait_dscnt 0" ::: "memory");

  // Sequential triangular part: lane = batch column.
#pragma unroll
  for (int jl = 0; jl < NLOC; ++jl) {
    const int j = JBASE + jl;
    float s = sh[aOff + j * A_STRIDE + lane] + sh[B_OFF + j];
#pragma unroll
    for (int il = 0; il < jl; ++il)  // contiguous uniform W reads (compiler merges)
      s = fmaf(sh[j * WT_STRIDE + JBASE + il], areg[il], s);
    const float aj = (j == N_NODES - 1) ? s : act_tanh(s);
    areg[jl] = aj;
    sh[aOff + j * A_STRIDE + lane] = aj;
  }
  asm volatile("s_wait_dscnt 0" ::: "memory");
}

__global__ __launch_bounds__(128) void dynnet_kernel(const float* __restrict__ x,
                                                     const float* __restrict__ W,
                                                     const float* __restrict__ bias,
                                                     float* __restrict__ out,
                                                     int nBatch, int nGroups) {
  __shared__ float sh[SH_FLOATS];
  const int tid  = threadIdx.x;
  const int wave = tid >> 5;
  const int lane = tid & 31;
  const int m    = lane & 15;
  const int h    = lane >> 4;
  const int aOff = A_OFF0 + wave * A_REGION;

  // Stage W transposed (Wt[j][i] = W[i][j]) + zero pad; stage bias.
  for (int t = tid; t < WT_ROWS * WT_STRIDE; t += 128) {
    const int j = t / WT_STRIDE, i = t % WT_STRIDE;
    sh[t] = (i < N_NODES && j < N_NODES) ? W[i * N_NODES + j] : 0.f;
  }
  for (int t = tid; t < N_NODES; t += 128) sh[B_OFF + t] = bias[t];
  __syncthreads();

  float areg[16];
  for (int g = blockIdx.x; g < nGroups; g += gridDim.x) {
    const int col = g * 128 + wave * 32 + lane;  // this lane's batch column
    const bool ok = col < nBatch;

    // Block 0: node 0 = input, nodes 1..15 purely intra-block.
    const float xv = ok ? x[col] : 0.f;
    areg[0] = xv;
    sh[aOff + lane] = xv;
#pragma unroll
    for (int jl = 1; jl < 16; ++jl) {
      float s = sh[B_OFF + jl];
#pragma unroll
      for (int il = 0; il < jl; ++il)
        s = fmaf(sh[jl * WT_STRIDE + il], areg[il], s);
      const float aj = act_tanh(s);
      areg[jl] = aj;
      sh[aOff + jl * A_STRIDE + lane] = aj;
    }
    asm volatile("s_wait_dscnt 0" ::: "memory");

    do_block<16, 16, 16>(sh, aOff, lane, m, h, areg);  //  4 K-chunks x 2 tiles
    do_block<32, 32, 16>(sh, aOff, lane, m, h, areg);  //  8 x 2
    do_block<48, 48, 16>(sh, aOff, lane, m, h, areg);  // 12 x 2
    do_block<64, 64,  2>(sh, aOff, lane, m, h, areg);  // 16 x 2; nodes 64 (tanh), 65 (linear)

    if (ok) out[col] = areg[1];  // node 65
  }
}

extern "C" void kernel_launch(void* const* d_in, const int* in_sizes, int n_in,
                              void* d_out, int out_size, void* d_ws, size_t ws_size,
                              hipStream_t stream) {
  const float* x = (const float*)d_in[0];   // [BATCH]
  const float* W = (const float*)d_in[1];   // [66*66] row-major
  const float* b = (const float*)d_in[2];   // [66]
  float* out = (float*)d_out;
  const int nBatch  = in_sizes[0];
  const int nGroups = (nBatch + 127) / 128;
  int grid = nGroups < 2048 ? nGroups : 2048;   // persistent-ish: amortize W staging
  dynnet_kernel<<<grid, 128, 0, stream>>>(x, W, b, out, nBatch, nGroups);
}